// NsaAttention_49993419325596
// MI455X (gfx1250) — compile-verified
//
#include <hip/hip_runtime.h>
#include <hip/hip_bf16.h>
#include <math.h>

// ---------------- NSA hyperparams / shapes (match reference) ----------------
constexpr int KS   = 32;
constexpr int ST   = 16;
constexpr int SEL  = 64;
constexpr int TOPN = 16;
constexpr int WIN  = 512;

constexpr int S   = 2048;
constexpr int Hq  = 16;
constexpr int Hk  = 2;
constexpr int G   = Hq / Hk;   // 8
constexpr int D   = 128;
constexpr int Dv  = 128;
constexpr int T   = (S - KS) / ST + 1;   // 127
constexpr int NSEL = S / SEL;            // 32

constexpr float NEGV  = -1e30f;
constexpr float SCALE = 0.08838834764831845f;   // 1/sqrt(128)

constexpr int LP = 132;   // padded LDS row pitch (floats) for staged tiles

typedef __attribute__((ext_vector_type(2))) float v2f;
typedef __attribute__((ext_vector_type(8))) float v8f;
typedef __attribute__((ext_vector_type(4))) int   v4i;

#define AS1 __attribute__((address_space(1)))
#define AS3 __attribute__((address_space(3)))

// fp32 WMMA: D(16x16,f32) = A(16x4,f32) * B(4x16,f32) + C
#define WMMA_F32(a, b, c) \
  __builtin_amdgcn_wmma_f32_16x16x4_f32(false, (a), false, (b), (short)0, (c), false, false)

// force "all DS-reads first, then the WMMA chain" issue order in a region
#if __has_builtin(__builtin_amdgcn_sched_group_barrier)
#define SCHED_DS_THEN_WMMA(nds, nwmma)                      \
  do {                                                      \
    __builtin_amdgcn_sched_group_barrier(0x100, (nds), 0);  \
    __builtin_amdgcn_sched_group_barrier(0x008, (nwmma), 0);\
  } while (0)
#else
#define SCHED_DS_THEN_WMMA(nds, nwmma) do {} while (0)
#endif

static __device__ __forceinline__ v2f mk2(float x, float y) { v2f r; r.x = x; r.y = y; return r; }

// async global->LDS 16B copy (gfx1250 GLOBAL_LOAD_ASYNC_TO_LDS_B128, ASYNCcnt)
static __device__ __forceinline__ void async_ld128(const float* g, float* l) {
#if __has_builtin(__builtin_amdgcn_global_load_async_to_lds_b128)
  __builtin_amdgcn_global_load_async_to_lds_b128((AS1 v4i*)g, (AS3 v4i*)l, 0, 0);
#else
  *(float4*)l = *(const float4*)g;   // sync fallback (still coalesced b128)
#endif
}

template <int N>
static __device__ __forceinline__ void wait_asynccnt() {
#if __has_builtin(__builtin_amdgcn_s_wait_asynccnt)
  __builtin_amdgcn_s_wait_asynccnt((unsigned short)N);
#else
  asm volatile("s_wait_asynccnt %0" :: "i"(N));
#endif
  asm volatile("" ::: "memory");
}

// reductions across the 16 lanes holding one C-matrix row (xor masks 1..8
// stay inside each 16-lane half of the wave32)
static __device__ __forceinline__ float rowmax16(float x) {
  #pragma unroll
  for (int off = 1; off < 16; off <<= 1) x = fmaxf(x, __shfl_xor(x, off, 32));
  return x;
}
static __device__ __forceinline__ float rowsum16(float x) {
  #pragma unroll
  for (int off = 1; off < 16; off <<= 1) x += __shfl_xor(x, off, 32);
  return x;
}

// ---------------- 1. block compression of K and V ----------------
__global__ void nsa_cmp_kv(const float* __restrict__ k, const float* __restrict__ v,
                           const float* __restrict__ wk, const float* __restrict__ wv,
                           float* __restrict__ cmpk, float* __restrict__ cmpv) {
  int idx = blockIdx.x * blockDim.x + threadIdx.x;     // over T*Hk*D
  if (idx >= T * Hk * D) return;
  int d = idx % D;
  int n = (idx / D) % Hk;
  int t = idx / (D * Hk);
  int base = t * ST;
  float sk = 0.f, sv = 0.f;
  #pragma unroll 8
  for (int j = 0; j < KS; ++j) {
    sk += k[((base + j) * Hk + n) * D  + d] * wk[j];
    sv += v[((base + j) * Hk + n) * Dv + d] * wv[j];
  }
  cmpk[idx] = sk;
  cmpv[idx] = sv;
}

__global__ void nsa_zero(float* __restrict__ p, int nelem) {
  int i = blockIdx.x * blockDim.x + threadIdx.x;
  if (i < nelem) p[i] = 0.f;
}

// ---------------- 2. compressed-KV attention + selection scores ----------------
// one wave per block; block = (16-query tile) x (query head)
__global__ void __launch_bounds__(32, 1)
nsa_cmp_attn(const float* __restrict__ q,
             const float* __restrict__ cmpk,
             const float* __restrict__ cmpv,
             float* __restrict__ p_slc,
             float* __restrict__ cmp_o) {
  __shared__ float sc[16][128];     // scores -> probabilities
  __shared__ float st[16][LP];      // staged cmpk / cmpv tile
  const int lane  = threadIdx.x;
  const int tile  = blockIdx.x & 127;
  const int head  = blockIdx.x >> 7;             // 0..15
  const int n     = head / G;                    // kv head
  const int qb    = tile * 16;
  const int row_a = lane & 15;                   // A-layout row
  const int kgrp  = (lane >> 4) * 2;             // A/B layout K sub-offset
  const int col   = lane & 15;                   // C-layout col / B col
  const int rhalf = (lane >> 4) * 8;             // C-layout row offset

  // preload q tile in WMMA-A layout: 32 chunks of K=4 over D=128
  v2f aq[32];
  #pragma unroll
  for (int c = 0; c < 32; ++c) {
    const float* qp = q + ((qb + row_a) * Hq + head) * D + c * 4 + kgrp;
    aq[c] = mk2(qp[0], qp[1]);
  }

  // scores against 128 (padded) compressed blocks
  for (int tb = 0; tb < 8; ++tb) {
    // stage cmpk rows [tb*16, tb*16+16) coalesced: one 512B row per issue
    #pragma unroll
    for (int i = 0; i < 16; ++i) {
      int tkey = tb * 16 + i;
      if (tkey < T) {
        async_ld128(cmpk + (tkey * Hk + n) * D + lane * 4, &st[i][lane * 4]);
      } else {
        #pragma unroll
        for (int w = 0; w < 4; ++w) st[i][lane * 4 + w] = 0.f;
      }
    }
    wait_asynccnt<0>();
    __syncthreads();

    // batch all B fragments, then run the WMMA chain back-to-back
    v2f bk[32];
    #pragma unroll
    for (int c = 0; c < 32; ++c)
      bk[c] = mk2(st[col][c * 4 + kgrp], st[col][c * 4 + kgrp + 1]);
    v8f c8 = {};
    #pragma unroll
    for (int c = 0; c < 32; ++c) c8 = WMMA_F32(aq[c], bk[c], c8);
    SCHED_DS_THEN_WMMA(16, 32);

    #pragma unroll
    for (int r = 0; r < 8; ++r) {
      int row  = r + rhalf;
      int t    = tb * 16 + col;
      int srow = qb + row;
      bool vis = (t < T) && ((t * ST + KS - 1) <= srow);
      sc[row][t] = vis ? c8[r] * SCALE : NEGV;
    }
    __syncthreads();
  }

  // per-row softmax (lanes 0..15 each own a row) + p_slc accumulation
  if (lane < 16) {
    int row = lane, srow = qb + row;
    float m = NEGV;
    for (int t = 0; t < 128; ++t) m = fmaxf(m, sc[row][t]);
    if (m > NEGV * 0.5f) {
      float sum = 0.f;
      for (int t = 0; t < 128; ++t) { float p = __expf(sc[row][t] - m); sum += p; sc[row][t] = p; }
      float inv = 1.f / sum;
      for (int t = 0; t < 128; ++t) sc[row][t] *= inv;
    } else {
      for (int t = 0; t < 128; ++t) sc[row][t] = 0.f;   // no visible block
    }
    // selection-block scores: block m receives p[t] for t in [4m-1, 4m+3]
    for (int m2 = 0; m2 < NSEL; ++m2) {
      int t0 = m2 * 4 - 1; if (t0 < 0) t0 = 0;
      int t1 = m2 * 4 + 3; if (t1 > T - 1) t1 = T - 1;
      float acc = 0.f;
      for (int t = t0; t <= t1; ++t) acc += sc[row][t];
      atomicAdd(&p_slc[(n * S + srow) * NSEL + m2], acc);
    }
  }
  __syncthreads();

  // PV: out[16][128] = P[16][128(pad)] @ cmp_v[n]
  v8f acc[8];
  #pragma unroll
  for (int nt = 0; nt < 8; ++nt) acc[nt] = (v8f){};

  for (int tc = 0; tc < 8; ++tc) {
    #pragma unroll
    for (int i = 0; i < 16; ++i) {
      int tkey = tc * 16 + i;
      if (tkey < T) {
        async_ld128(cmpv + (tkey * Hk + n) * Dv + lane * 4, &st[i][lane * 4]);
      } else {
        #pragma unroll
        for (int w = 0; w < 4; ++w) st[i][lane * 4 + w] = 0.f;
      }
    }
    wait_asynccnt<0>();
    __syncthreads();

    // A fragments for this 16-wide K slab (from probability matrix in LDS)
    v2f a4[4];
    #pragma unroll
    for (int kb = 0; kb < 16; kb += 4)
      a4[kb >> 2] = mk2(sc[row_a][tc * 16 + kb + kgrp], sc[row_a][tc * 16 + kb + kgrp + 1]);

    #pragma unroll
    for (int kb = 0; kb < 16; kb += 4) {
      v2f bv[8];
      #pragma unroll
      for (int nt = 0; nt < 8; ++nt)
        bv[nt] = mk2(st[kb + kgrp][nt * 16 + col], st[kb + kgrp + 1][nt * 16 + col]);
      #pragma unroll
      for (int nt = 0; nt < 8; ++nt)
        acc[nt] = WMMA_F32(a4[kb >> 2], bv[nt], acc[nt]);
    }
    SCHED_DS_THEN_WMMA(24, 32);
    __syncthreads();
  }

  #pragma unroll
  for (int nt = 0; nt < 8; ++nt)
    #pragma unroll
    for (int r = 0; r < 8; ++r)
      cmp_o[((qb + r + rhalf) * Hq + head) * Dv + nt * 16 + col] = acc[nt][r];
}

// ---------------- 3. top-n selection -> per-row block bitmask ----------------
__global__ void nsa_topk(const float* __restrict__ p_slc, unsigned int* __restrict__ selmask) {
  int idx = blockIdx.x * blockDim.x + threadIdx.x;   // over Hk*S
  if (idx >= Hk * S) return;
  int s = idx % S, n = idx / S;
  int cur = s / SEL;
  float sc[NSEL];
  #pragma unroll
  for (int m = 0; m < NSEL; ++m) {
    bool causal = (m * SEL) <= s;
    bool forced = (m == 0) || (m == cur);
    float v = p_slc[(n * S + s) * NSEL + m];
    sc[m] = !causal ? NEGV : (forced ? 1e30f : v);
  }
  unsigned int mask = 0;
  #pragma unroll
  for (int m = 0; m < NSEL; ++m) {
    if (sc[m] <= NEGV * 0.5f) continue;            // "valid" check
    int rank = 0;
    #pragma unroll
    for (int j = 0; j < NSEL; ++j)
      if (sc[j] > sc[m] || (sc[j] == sc[m] && j < m)) rank++;
    if (rank < TOPN) mask |= (1u << m);
  }
  selmask[idx] = mask;
}

// ---------------- 4. selected + windowed attention, gated combine ----------------
// one wave per block; block = (16-query tile) x (query head). Flash-style
// streaming over causal key tiles with block skipping; double-buffered async
// global->LDS staging of K/V tiles; two online-softmax accumulators
// (window / selection) sharing one WMMA score tile.
__global__ void __launch_bounds__(32, 1)
nsa_main_attn(const float* __restrict__ q, const float* __restrict__ k,
              const float* __restrict__ v,
              const float* __restrict__ w_gate, const float* __restrict__ b_gate,
              const unsigned int* __restrict__ selmask,
              const float* __restrict__ cmp_o,
              float* __restrict__ out) {
  __shared__ float kt[2][16][LP];
  __shared__ float vt[2][16][LP];
  __shared__ float p_lds[16][17];
  __shared__ float gate_lds[16][3];
  __shared__ unsigned int smask_lds[16];
  const int lane  = threadIdx.x;
  const int tile  = blockIdx.x & 127;
  const int head  = blockIdx.x >> 7;
  const int n     = head / G;
  const int qb    = tile * 16;
  const int row_a = lane & 15;
  const int kgrp  = (lane >> 4) * 2;
  const int col   = lane & 15;
  const int rhalf = (lane >> 4) * 8;
  const float INFF = __builtin_inff();

  // per-row selection masks (+ union for tile skipping)
  unsigned int rmask = selmask[n * S + qb + (lane & 15)];
  unsigned int un = rmask;
  #pragma unroll
  for (int off = 1; off < 16; off <<= 1) un |= __shfl_xor(un, off, 32);
  if (lane < 16) smask_lds[lane] = rmask;

  // sigmoid gates, one row per lane 0..15
  if (lane < 16) {
    const float* qp = q + ((qb + lane) * Hq + head) * D;
    float g0 = b_gate[0], g1 = b_gate[1], g2 = b_gate[2];
    for (int d = 0; d < D; ++d) {
      float qv = qp[d];
      g0 += qv * w_gate[d * 3 + 0];
      g1 += qv * w_gate[d * 3 + 1];
      g2 += qv * w_gate[d * 3 + 2];
    }
    gate_lds[lane][0] = 1.f / (1.f + __expf(-g0));
    gate_lds[lane][1] = 1.f / (1.f + __expf(-g1));
    gate_lds[lane][2] = 1.f / (1.f + __expf(-g2));
  }
  __syncthreads();

  // preload q tile (A layout)
  v2f aq[32];
  #pragma unroll
  for (int c = 0; c < 32; ++c) {
    const float* qp = q + ((qb + row_a) * Hq + head) * D + c * 4 + kgrp;
    aq[c] = mk2(qp[0], qp[1]);
  }

  v8f accw[8], accs[8];
  float mw[8], lw[8], ms[8], ls[8];
  #pragma unroll
  for (int r = 0; r < 8; ++r) {
    accw[r] = (v8f){}; accs[r] = (v8f){};
    mw[r] = -INFF; lw[r] = 0.f; ms[r] = -INFF; ls[r] = 0.f;
  }

  auto tile_active = [&](int jt) -> bool {
    int jb = jt * 16;
    return ((jb + 15) >= (qb - WIN)) || ((un >> (jb >> 6)) & 1u);
  };
  // coalesced async staging of one K tile + one V tile (32 x b128 row loads)
  auto stage_tile = [&](int jt, int buf) {
    const int jb = jt * 16;
    #pragma unroll
    for (int i = 0; i < 16; ++i)
      async_ld128(k + ((jb + i) * Hk + n) * D  + lane * 4, &kt[buf][i][lane * 4]);
    #pragma unroll
    for (int i = 0; i < 16; ++i)
      async_ld128(v + ((jb + i) * Hk + n) * Dv + lane * 4, &vt[buf][i][lane * 4]);
  };

  int jt = 0;
  while (jt <= tile && !tile_active(jt)) ++jt;   // jt==tile is always window-active
  int buf = 0;
  stage_tile(jt, buf);

  while (true) {
    int jn = jt + 1;
    while (jn <= tile && !tile_active(jn)) ++jn;
    const bool have_next = (jn <= tile);
    if (have_next) stage_tile(jn, buf ^ 1);      // prefetch next active tile
    if (have_next) wait_asynccnt<32>();          // current buffer complete
    else           wait_asynccnt<0>();
    __syncthreads();

    const int jb = jt * 16;
    const int selblk = jb >> 6;
    const bool winAct = (jb + 15) >= (qb - WIN);
    const bool selAct = (un >> selblk) & 1u;

    // shared 16x16 score tile: batch all B fragments from LDS, then run
    // the 32-deep fp32 WMMA chain back-to-back
    v2f bk[32];
    #pragma unroll
    for (int c = 0; c < 32; ++c)
      bk[c] = mk2(kt[buf][col][c * 4 + kgrp], kt[buf][col][c * 4 + kgrp + 1]);
    v8f s8 = {};
    #pragma unroll
    for (int c = 0; c < 32; ++c) s8 = WMMA_F32(aq[c], bk[c], s8);
    SCHED_DS_THEN_WMMA(16, 32);

    auto branch = [&](bool is_win, float (&mm)[8], float (&ll)[8], v8f (&acc)[8]) {
      float pr[8], al[8];
      #pragma unroll
      for (int r = 0; r < 8; ++r) {
        int srow = qb + r + rhalf;
        int j = jb + col;
        bool ok = (j <= srow);
        if (is_win) ok = ok && ((srow - j) <= WIN);
        else        ok = ok && ((smask_lds[r + rhalf] >> selblk) & 1u);
        float x  = ok ? s8[r] * SCALE : -INFF;
        float rm = rowmax16(x);
        float mn = fmaxf(mm[r], rm);
        float a  = (mm[r] == -INFF) ? 0.f : __expf(mm[r] - mn);
        float p  = ok ? __expf(x - mn) : 0.f;
        ll[r] = ll[r] * a + rowsum16(p);
        mm[r] = mn;
        al[r] = a; pr[r] = p;
      }
      #pragma unroll
      for (int nt = 0; nt < 8; ++nt)
        #pragma unroll
        for (int r = 0; r < 8; ++r) acc[nt][r] *= al[r];
      __syncthreads();
      #pragma unroll
      for (int r = 0; r < 8; ++r) p_lds[r + rhalf][col] = pr[r];  // C layout -> LDS
      __syncthreads();
      // P(A layout) @ V ; A fragments batched once, B fragments batched per
      // K-chunk so the 8 independent acc chains run back-to-back
      v2f a4[4];
      #pragma unroll
      for (int kb = 0; kb < 16; kb += 4)
        a4[kb >> 2] = mk2(p_lds[row_a][kb + kgrp], p_lds[row_a][kb + kgrp + 1]);
      #pragma unroll
      for (int kb = 0; kb < 16; kb += 4) {
        v2f bv[8];
        #pragma unroll
        for (int nt = 0; nt < 8; ++nt)
          bv[nt] = mk2(vt[buf][kb + kgrp][nt * 16 + col],
                       vt[buf][kb + kgrp + 1][nt * 16 + col]);
        #pragma unroll
        for (int nt = 0; nt < 8; ++nt)
          acc[nt] = WMMA_F32(a4[kb >> 2], bv[nt], acc[nt]);
      }
      SCHED_DS_THEN_WMMA(24, 32);
    };
    if (winAct) branch(true,  mw, lw, accw);
    if (selAct) branch(false, ms, ls, accs);

    if (!have_next) break;
    jt = jn; buf ^= 1;
  }

  // gated combine with compressed-attention output
  #pragma unroll
  for (int nt = 0; nt < 8; ++nt) {
    #pragma unroll
    for (int r = 0; r < 8; ++r) {
      int row  = r + rhalf;
      int srow = qb + row;
      float g0 = gate_lds[row][0], g1 = gate_lds[row][1], g2 = gate_lds[row][2];
      float so = (ls[r] > 0.f) ? accs[nt][r] / ls[r] : 0.f;
      float wo = (lw[r] > 0.f) ? accw[nt][r] / lw[r] : 0.f;
      float co = cmp_o[(srow * Hq + head) * Dv + nt * 16 + col];
      out[(srow * Hq + head) * Dv + nt * 16 + col] = g0 * co + g1 * so + g2 * wo;
    }
  }
}

// ---------------- launch ----------------
extern "C" void kernel_launch(void* const* d_in, const int* in_sizes, int n_in,
                              void* d_out, int out_size, void* d_ws, size_t ws_size,
                              hipStream_t stream) {
  const float* q   = (const float*)d_in[0];
  const float* k   = (const float*)d_in[1];
  const float* v   = (const float*)d_in[2];
  const float* wck = (const float*)d_in[3];
  const float* wcv = (const float*)d_in[4];
  const float* wg  = (const float*)d_in[5];
  const float* bg  = (const float*)d_in[6];
  float* out = (float*)d_out;

  // workspace partition (floats): cmpk | cmpv | p_slc | selmask | cmp_o
  float* ws    = (float*)d_ws;
  float* cmpk  = ws;
  float* cmpv  = cmpk + T * Hk * D;
  float* p_slc = cmpv + T * Hk * Dv;
  unsigned int* selmask = (unsigned int*)(p_slc + Hk * S * NSEL);
  float* cmp_o = (float*)(selmask + Hk * S);

  nsa_cmp_kv<<<(T * Hk * D + 255) / 256, 256, 0, stream>>>(k, v, wck, wcv, cmpk, cmpv);
  nsa_zero<<<(Hk * S * NSEL + 255) / 256, 256, 0, stream>>>(p_slc, Hk * S * NSEL);
  nsa_cmp_attn<<<128 * Hq, 32, 0, stream>>>(q, cmpk, cmpv, p_slc, cmp_o);
  nsa_topk<<<(Hk * S + 255) / 256, 256, 0, stream>>>(p_slc, selmask);
  nsa_main_attn<<<128 * Hq, 32, 0, stream>>>(q, k, v, wg, bg, selmask, cmp_o, out);
}